// ROIpooling_27814208209247
// MI455X (gfx1250) — compile-verified
//
#include <hip/hip_runtime.h>
#include <stdint.h>
#include <float.h>

// ROI max-pool, MI455X (gfx1250).
// features: [1, 512, 64, 64] f32, rois: [1024, 4] i32 -> out: [1024, 512, 7, 7] f32
// Write-bandwidth bound (~103 MB stores @ 23.3 TB/s ~ 4.4 us). Feature map (8 MB)
// is L2-resident. Per-ROI window <= 9x9 feature cells.
// Staging: CDNA5 Tensor Data Mover, one tensor_load_to_lds per 64-channel chunk,
// DOUBLE-BUFFERED: TDM ops from one wave complete in-order, so wave0 issues chunk
// k+1, waits tensorcnt<=1 (chunk k landed, k+1 in flight), and the block computes
// chunk k while the DMA streams chunk k+1. No WMMA: max-pool has no MAC structure.

#define TPB    256
#define CB     64       // channels per LDS chunk (x2 buffers = 41.5 KB LDS)
#define NCH    512
#define NCHUNK (NCH / CB)
#define FH     64
#define FW     64
#define MAXWIN 9        // max ROI window extent in feature cells

typedef unsigned int u32;
typedef u32 v4u __attribute__((ext_vector_type(4)));
typedef int v8i __attribute__((ext_vector_type(8)));
typedef int v4i __attribute__((ext_vector_type(4)));

#if defined(__HIP_DEVICE_COMPILE__) && \
    __has_builtin(__builtin_amdgcn_tensor_load_to_lds) && \
    __has_builtin(__builtin_amdgcn_s_wait_tensorcnt)
#define USE_TDM 1
#else
#define USE_TDM 0
#endif

#if defined(__HIP_DEVICE_COMPILE__) && \
    __has_builtin(__builtin_amdgcn_global_load_async_to_lds_b32) && \
    __has_builtin(__builtin_amdgcn_s_wait_asynccnt)
#define USE_ASYNC 1
#else
#define USE_ASYNC 0
#endif

__global__ __launch_bounds__(TPB) void roi_pool_kernel(
    const float* __restrict__ fmap,   // [512, 64, 64]
    const int*   __restrict__ rois,   // [R, 4] image-pixel (x1,y1,x2,y2)
    float* __restrict__ out)          // [R, 512, 7, 7]
{
  // Two packed window tiles: [CB][hlen][wlen], worst case CB*9*9 floats each.
  __shared__ float tile[2][CB * MAXWIN * MAXWIN];

  const int r   = blockIdx.x;
  const int tid = threadIdx.x;

  // rois.float()*(1/16) truncated == >>4 for non-negative coords (exact pow2 mul).
  const int x1 = rois[4 * r + 0] >> 4;
  const int y1 = rois[4 * r + 1] >> 4;
  const int x2 = rois[4 * r + 2] >> 4;
  const int y2 = rois[4 * r + 3] >> 4;
  const u32 wlen = (u32)(x2 - x1 + 1);   // 1..9
  const u32 hlen = (u32)(y2 - y1 + 1);   // 1..9
  const u32 hw   = wlen * hlen;

  // Issue staging of channel chunk `ck` into buffer `buf`.
  auto stage = [&](int ck, int buf) {
    const int c0 = ck * CB;
#if USE_TDM
    if (tid < 32) {   // one wave owns the tensor DMA
      const uint64_t gaddr =
          (uint64_t)(uintptr_t)(fmap + (((size_t)c0 * FH + (u32)y1) * FW + (u32)x1));
      const u32 laddr = (u32)(uintptr_t)&tile[buf][0]; // generic low 32b = LDS addr

      v4u g0 = {};                       // D# group 0 (128b)
      g0[0] = 1u;                        // count=1 (valid), user mode, no gather
      g0[1] = laddr;                     // lds_addr
      g0[2] = (u32)(gaddr & 0xFFFFFFFFu);            // global_addr[31:0]
      g0[3] = (u32)((gaddr >> 32) & 0x01FFFFFFu)     // global_addr[56:32]
            | (2u << 30);                             // type=2 ("image")

      v8i g1 = {};                       // D# group 1 (256b)
      g1[0] = (int)(2u << 16);           // data_size=2 -> 4B; mask=0; no pad/iter
      g1[1] = (int)(wlen << 16);         // tensor_dim0[15:0] @ bits[63:48]
      g1[2] = (int)(hlen << 16);         // tensor_dim0 hi=0 | tensor_dim1[15:0]
      g1[3] = (int)(wlen << 16);         // tensor_dim1 hi=0 | tile_dim0=wlen
      g1[4] = (int)(hlen | ((u32)CB << 16));      // tile_dim1=hlen, tile_dim2=CB
      g1[5] = (int)FW;                   // tensor_dim0_stride = 64 elems (row)
      g1[6] = (int)((u32)(FH * FW) << 16);        // stride0 hi=0 | stride1[15:0]
      g1[7] = (int)((u32)(FH * FW) >> 16);        // stride1[47:16] = 0

      v4i g2 = {};                       // D# group 2
      g2[0] = CB;                        // tensor_dim2 = CB (tile==tensor, no OOB)
      g2[1] = 1;                         // tensor_dim3 (unused; tile_dim3=0)
      v4i g3 = {};                       // D# group 3 (unused dims)

#if __clang_major__ >= 23
      v8i gx = {};                       // extra arg in clang-23 6-arg form
      __builtin_amdgcn_tensor_load_to_lds(g0, g1, g2, g3, gx, 0);
#else
      __builtin_amdgcn_tensor_load_to_lds(g0, g1, g2, g3, 0);
#endif
    }
#else
    // Fallback staging: per-lane copies into the same packed LDS layout.
    const u32 nelem = (u32)CB * hw;
    for (u32 idx = (u32)tid; idx < nelem; idx += TPB) {
      u32 cc  = idx / hw;
      u32 rem = idx - cc * hw;
      u32 dy  = rem / wlen;
      u32 dx  = rem - dy * wlen;
      const float* g =
          fmap + (((size_t)((u32)c0 + cc) * FH + ((u32)y1 + dy)) * FW + ((u32)x1 + dx));
      float* l = &tile[buf][idx];
#if USE_ASYNC
      __builtin_amdgcn_global_load_async_to_lds_b32(
          (__attribute__((address_space(1))) void*)(uintptr_t)g,
          (__attribute__((address_space(3))) void*)(u32)(uintptr_t)l, 0, 0);
#else
      *l = *g;
#endif
    }
#if USE_ASYNC
    __builtin_amdgcn_s_wait_asynccnt(0);   // fallback is synchronous
#endif
#endif
  };

  stage(0, 0);                         // prologue: fill buffer 0

  for (int k = 0; k < NCHUNK; ++k) {
    if (k + 1 < NCHUNK) stage(k + 1, (k + 1) & 1);   // next DMA in flight
#if USE_TDM
    if (tid < 32) {
      // In-order TDM completion: <=1 outstanding leaves chunk k complete while
      // chunk k+1 streams; last chunk drains to 0.
      if (k + 1 < NCHUNK) __builtin_amdgcn_s_wait_tensorcnt(1);
      else                __builtin_amdgcn_s_wait_tensorcnt(0);
    }
#endif
    __syncthreads();                   // chunk k visible to all waves

    // ---- 7x7 adaptive max over staged chunk k; coalesced b32 stores.
    const int c0 = k * CB;
    const float* tbase = &tile[k & 1][0];
    for (u32 o = (u32)tid; o < (u32)CB * 49u; o += TPB) {
      u32 cc = o / 49u;
      u32 p  = o - cc * 49u;
      u32 ph = p / 7u;
      u32 pw = p - ph * 7u;
      // bin i covers [floor(i*L/7), ceil((i+1)*L/7)) -- never empty for L>=1
      u32 hs = (ph * hlen) / 7u;
      u32 he = ((ph + 1u) * hlen + 6u) / 7u;
      u32 ws = (pw * wlen) / 7u;
      u32 we = ((pw + 1u) * wlen + 6u) / 7u;
      const float* base = tbase + cc * hw;
      float m = -FLT_MAX;               // == jnp.finfo(float32).min
      for (u32 dy = hs; dy < he; ++dy)
        for (u32 dx = ws; dx < we; ++dx)
          m = fmaxf(m, base[dy * wlen + dx]);
      out[((size_t)r * NCH + (u32)c0 + cc) * 49u + p] = m;
    }
    __syncthreads();   // all reads of tile[k&1] done before it is re-filled
  }
}

extern "C" void kernel_launch(void* const* d_in, const int* in_sizes, int n_in,
                              void* d_out, int out_size, void* d_ws, size_t ws_size,
                              hipStream_t stream) {
  const float* fmap = (const float*)d_in[0];
  const int*   rois = (const int*)d_in[1];
  const int R = in_sizes[1] / 4;   // 1024
  roi_pool_kernel<<<dim3((unsigned)R), dim3(TPB), 0, stream>>>(fmap, rois, (float*)d_out);
  (void)n_in; (void)out_size; (void)d_ws; (void)ws_size;
}